// PointsToObjects_1511828488715
// MI455X (gfx1250) — compile-verified
//
#include <hip/hip_runtime.h>
#include <cstdint>
#include <cstddef>

// Problem constants (from reference: (16, 24, 256, 256) f32, TOP_K=42, MIN_CONF=0.1)
#define B_    16
#define CTOT  24
#define CH    20               // heat channels
#define HW    65536            // 256*256
#define NHEAT (CH * HW)        // 1,310,720 per batch
#define K_    42
#define PB    32               // phase-1 blocks per batch
#define T1    128              // phase-1 threads per block (4 waves)
#define SLICE (NHEAT / PB)     // 40,960
#define CHUNK 2048             // floats per LDS tile (8 KB)
#define NCHUNK (SLICE / CHUNK) // 20
#define MPT   (CHUNK / T1)     // 16 elements per thread per chunk
#define NCAND (PB * K_)        // 1344 candidates per batch
#define SORTN 2048
#define T2    256

// Priority: matches jax.lax.top_k ordering (value desc, index asc on ties)
__device__ __forceinline__ bool pgt(float v1, int i1, float v2, int i2) {
    return (v1 > v2) || ((v1 == v2) && (i1 < i2));
}

// ---------------------------------------------------------------------------
// Phase 1: per-slice exact top-42 using async global->LDS streaming (CDNA5 TDM
// async path: global_load_async_to_lds_b128 + s_wait_asynccnt, double buffered)
// ---------------------------------------------------------------------------
__global__ __launch_bounds__(T1) void p1_topk_slices(const float* __restrict__ in,
                                                     float* __restrict__ wsV,
                                                     int*   __restrict__ wsI) {
    __shared__ __align__(16) float tile[2][CHUNK];   // 16 KB double buffer
    __shared__ float lv[K_ * T1];                    // per-thread sorted lists (strided)
    __shared__ int   li[K_ * T1];
    __shared__ float rv[T1];
    __shared__ int   ri[T1];
    __shared__ int   rt[T1];
    __shared__ int   winner;

    const int tid = threadIdx.x;
    const int bid = blockIdx.x;          // 0..511
    const int b   = bid / PB;
    const int p   = bid % PB;
    const int sliceStart = p * SLICE;    // element index within this batch's heat
    const float* gbase = in + (size_t)b * (CTOT * HW) + sliceStart;

    // ---- issue async DMA for chunk 0 into buffer 0 ----
    {
        unsigned lbase = (unsigned)(unsigned long long)(void*)&tile[0][0];
        const char* g = (const char*)gbase;
#pragma unroll
        for (int r = 0; r < 4; ++r) {
            unsigned boff = (unsigned)((r * T1 + tid) * 16);
            asm volatile("global_load_async_to_lds_b128 %0, %1, off"
                         :: "v"(lbase + boff),
                            "v"((unsigned long long)(g + boff))
                         : "memory");
        }
    }

    int cnt = 0;                         // entries in this thread's list
    for (int c = 0; c < NCHUNK; ++c) {
        const int cur = c & 1;
        if (c + 1 < NCHUNK) {
            // prefetch next chunk into other buffer, then wait only for current
            unsigned lbase = (unsigned)(unsigned long long)(void*)&tile[cur ^ 1][0];
            const char* g = (const char*)(gbase + (size_t)(c + 1) * CHUNK);
#pragma unroll
            for (int r = 0; r < 4; ++r) {
                unsigned boff = (unsigned)((r * T1 + tid) * 16);
                asm volatile("global_load_async_to_lds_b128 %0, %1, off"
                             :: "v"(lbase + boff),
                                "v"((unsigned long long)(g + boff))
                             : "memory");
            }
            // 4 outstanding ops belong to chunk c+1; async loads complete in order,
            // so <=4 outstanding means chunk c has fully landed in LDS.
            asm volatile("s_wait_asynccnt 4" ::: "memory");
        } else {
            asm volatile("s_wait_asynccnt 0" ::: "memory");
        }
        __syncthreads();   // all 4 waves' DMA visible to everyone

        const int idxBase = sliceStart + c * CHUNK;
#pragma unroll 4
        for (int m = 0; m < MPT; ++m) {
            const int off = tid + T1 * m;
            const float v = tile[cur][off];
            const int idx = idxBase + off;
            if (cnt == K_) {
                // replace-min path: list ascending, slot 0 = current min
                if (pgt(v, idx, lv[tid], li[tid])) {
                    int s = 1;
                    while (s < K_) {
                        float sv2 = lv[s * T1 + tid]; int sj = li[s * T1 + tid];
                        if (pgt(v, idx, sv2, sj)) {
                            lv[(s - 1) * T1 + tid] = sv2; li[(s - 1) * T1 + tid] = sj; ++s;
                        } else break;
                    }
                    lv[(s - 1) * T1 + tid] = v; li[(s - 1) * T1 + tid] = idx;
                }
            } else {
                // warm-up insertion keeping ascending order
                int s = cnt;
                while (s > 0) {
                    float sv2 = lv[(s - 1) * T1 + tid]; int sj = li[(s - 1) * T1 + tid];
                    if (pgt(sv2, sj, v, idx)) {
                        lv[s * T1 + tid] = sv2; li[s * T1 + tid] = sj; --s;
                    } else break;
                }
                lv[s * T1 + tid] = v; li[s * T1 + tid] = idx; ++cnt;
            }
        }
        __syncthreads();   // everyone done reading tile[cur] before it is refilled
    }

    // ---- k-way merge of 128 sorted lists: pop global max 42 times ----
    int hp = cnt - 1;                    // head pointer (max end of ascending list)
    for (int r = 0; r < K_; ++r) {
        if (hp >= 0) { rv[tid] = lv[hp * T1 + tid]; ri[tid] = li[hp * T1 + tid]; }
        else         { rv[tid] = -__builtin_inff(); ri[tid] = 0x7fffffff; }
        rt[tid] = tid;
        __syncthreads();
        for (int offr = T1 / 2; offr > 0; offr >>= 1) {
            if (tid < offr && pgt(rv[tid + offr], ri[tid + offr], rv[tid], ri[tid])) {
                rv[tid] = rv[tid + offr]; ri[tid] = ri[tid + offr]; rt[tid] = rt[tid + offr];
            }
            __syncthreads();
        }
        if (tid == 0) {
            wsV[bid * K_ + r] = rv[0];
            wsI[bid * K_ + r] = ri[0];
            winner = rt[0];
        }
        __syncthreads();
        if (tid == winner) --hp;
        __syncthreads();
    }
}

// ---------------------------------------------------------------------------
// Phase 2: per-batch reduce 1344 candidates -> sorted top-42, gather tail, pack
// ---------------------------------------------------------------------------
__global__ __launch_bounds__(T2) void p2_final(const float* __restrict__ in,
                                               const float* __restrict__ wsV,
                                               const int*   __restrict__ wsI,
                                               float* __restrict__ out) {
    __shared__ float sv[SORTN];
    __shared__ int   si[SORTN];
    const int tid = threadIdx.x;
    const int b   = blockIdx.x;

    for (int i = tid; i < SORTN; i += T2) {
        if (i < NCAND) { sv[i] = wsV[b * NCAND + i]; si[i] = wsI[b * NCAND + i]; }
        else           { sv[i] = -__builtin_inff(); si[i] = 0x7fffffff; }
    }
    __syncthreads();

    // bitonic sort, descending by priority (value desc, index asc on ties)
    for (int k = 2; k <= SORTN; k <<= 1) {
        for (int j = k >> 1; j > 0; j >>= 1) {
            for (int i = tid; i < SORTN; i += T2) {
                const int l = i ^ j;
                if (l > i) {
                    float a = sv[i], c2 = sv[l];
                    int  ia = si[i], ic = si[l];
                    const bool desc = ((i & k) == 0);
                    const bool sw = desc ? pgt(c2, ic, a, ia) : pgt(a, ia, c2, ic);
                    if (sw) { sv[i] = c2; sv[l] = a; si[i] = ic; si[l] = ia; }
                }
            }
            __syncthreads();
        }
    }

    if (tid < K_) {
        const float score = sv[tid];
        const int idx = si[tid];               // in [0, 20*65536)
        const int cls = idx >> 16;             // / HW
        const int rem = idx & 0xFFFF;          // % HW
        const int ys  = rem >> 8;
        const int xs  = rem & 255;
        const float* tb = in + (size_t)b * (CTOT * HW) + (size_t)CH * HW;
        const float dy = tb[rem];
        const float dx = tb[HW + rem];
        const float hh = tb[2 * HW + rem];
        const float ww = tb[3 * HW + rem];
        const float keep = (score > 0.1f) ? 1.0f : 0.0f;
        float* o = out + ((size_t)b * K_ + tid) * 6;
        o[0] = ((float)ys + dy) * keep;
        o[1] = ((float)xs + dx) * keep;
        o[2] = hh * keep;
        o[3] = ww * keep;
        o[4] = (float)cls * keep;
        o[5] = score * keep;
    }
}

extern "C" void kernel_launch(void* const* d_in, const int* in_sizes, int n_in,
                              void* d_out, int out_size, void* d_ws, size_t ws_size,
                              hipStream_t stream) {
    (void)in_sizes; (void)n_in; (void)out_size; (void)ws_size;
    const float* in = (const float*)d_in[0];
    // workspace: 512*42 floats (86,016 B) then 512*42 ints (86,016 B) = 172 KB
    float* wsV = (float*)d_ws;
    int*   wsI = (int*)((char*)d_ws + (size_t)(B_ * PB * K_) * sizeof(float));

    p1_topk_slices<<<dim3(B_ * PB), dim3(T1), 0, stream>>>(in, wsV, wsI);
    p2_final<<<dim3(B_), dim3(T2), 0, stream>>>(in, wsV, wsI, (float*)d_out);
}